// _RoIRingPooling_65901978190518
// MI455X (gfx1250) — compile-verified
//
#include <hip/hip_runtime.h>
#include <stdint.h>

#define N_IMG   4
#define C_DIM   256
#define H_DIM   50
#define W_DIM   50
#define R_DIM   128
#define PH      7
#define PW      7
#define SP_SCALE 0.0625f
#define CH      4            // channels staged per block

typedef unsigned int uint32;
typedef uint32 v4u  __attribute__((ext_vector_type(4)));
typedef int    v8i  __attribute__((ext_vector_type(8)));
typedef int    v4i  __attribute__((ext_vector_type(4)));

static __device__ __forceinline__ int imin(int a, int b) { return a < b ? a : b; }
static __device__ __forceinline__ int imax(int a, int b) { return a > b ? a : b; }
static __device__ __forceinline__ int rfl(int x) { return __builtin_amdgcn_readfirstlane(x); }

__global__ __launch_bounds__(256) void roi_ring_pool_kernel(
    const float* __restrict__ feat,   // [N, C, H, W]
    const float* __restrict__ rois,   // [R, 5]
    float* __restrict__ out)          // [R, C, 7, 7]
{
    __shared__ float s_tile[CH * H_DIM * W_DIM];   // 40 KB: CH channel row-spans

    const int r   = blockIdx.x;          // ROI index
    const int c0  = blockIdx.y * CH;     // first channel of this block
    const int tid = threadIdx.x;

    // ---------------- per-ROI geometry (block-uniform) ----------------
    const float* roi = rois + r * 5;
    int   b  = (int)roi[0];
    float x1 = roi[1], y1 = roi[2], x2 = roi[3], y2 = roi[4];
    float rw = x2 - x1 + 1.0f, rh = y2 - y1 + 1.0f;
    float xc = (x1 + x2) * 0.5f, yc = (y1 + y2) * 0.5f;

    // outer box, scale 1.8  (jnp.round == rintf, RNE)
    float swO = rw * (1.8f * 0.5f), shO = rh * (1.8f * 0.5f);
    int ox1 = (int)rintf((xc - swO) * SP_SCALE);
    int oy1 = (int)rintf((yc - shO) * SP_SCALE);
    int ox2 = (int)rintf((xc + swO) * SP_SCALE);
    int oy2 = (int)rintf((yc + shO) * SP_SCALE);
    // inner box, scale 1.0
    float swI = rw * 0.5f, shI = rh * 0.5f;
    int ix1 = (int)rintf((xc - swI) * SP_SCALE);
    int iy1 = (int)rintf((yc - shI) * SP_SCALE);
    int ix2 = (int)rintf((xc + swI) * SP_SCALE);
    int iy2 = (int)rintf((yc + shI) * SP_SCALE);

    int roiW = imax(ox2 - ox1 + 1, 1);
    int roiH = imax(oy2 - oy1 + 1, 1);
    float bsh = (float)roiH / (float)PH;
    float bsw = (float)roiW / (float)PW;

    // union of all bin rows: [h0, h1)  (rows are contiguous in memory)
    int h0 = imin(imax(oy1, 0), H_DIM);                                    // hstart[0]
    int h1 = imin(imax((int)ceilf((float)PH * bsh) + oy1, 0), H_DIM);      // hend[PH-1]
    int nRows  = imax(h1 - h0, 0);
    int nElems = nRows * W_DIM;          // per-channel tile size (<= 2500)

    // ---------------- TDM stage: global -> LDS (wave 0 only) ----------------
    int b_rf  = rfl(b);
    int h0_rf = rfl(h0);
    int ne_rf = rfl(nElems);
    uint64_t ga = (uint64_t)(uintptr_t)feat +
                  4ull * (((uint64_t)(b_rf * C_DIM + c0) * H_DIM + (uint64_t)h0_rf) * W_DIM);
    uint32 ldsA = (uint32)(uintptr_t)(void*)&s_tile[0];   // flat LDS addr truncates to LDS offset

    if (tid < 32) {
        uint32 ne      = (uint32)ne_rf;
        uint32 stride0 = (uint32)(H_DIM * W_DIM);   // plane-to-plane stride (elements)

        // D# group 0 (128b): count=1, lds_addr, 57b global_addr, type=2
        v4u g0;
        g0.x = 1u;                                             // count=1, user desc
        g0.y = ldsA;                                           // lds_addr[31:0]
        g0.z = (uint32)ga;                                     // global_addr[31:0]
        g0.w = ((uint32)(ga >> 32) & 0x1FFFFFFu) | (2u << 30); // global_addr[56:32] | type=2

        // D# group 1 (256b): data_size=4B(code 2), tensor_dim0=ne, tensor_dim1=CH,
        // tile_dim0=ne, tile_dim1=CH, tile_dim2=0, dim0_stride=H*W, dim1_stride=0
        v8i g1;
        g1[0] = (int)(2u << 16);                               // data_size = 4 bytes
        g1[1] = (int)((ne & 0xFFFFu) << 16);                   // tensor_dim0[15:0]
        g1[2] = (int)(((ne >> 16) & 0xFFFFu) | ((uint32)CH << 16)); // tdim0[31:16] | tdim1[15:0]
        g1[3] = (int)((ne & 0xFFFFu) << 16);                   // tdim1[31:16]=0 | tile_dim0
        g1[4] = (int)CH;                                       // tile_dim1 | tile_dim2=0
        g1[5] = (int)stride0;                                  // tensor_dim0_stride[31:0]
        g1[6] = 0;                                             // stride0[47:32] | stride1[15:0]
        g1[7] = 0;                                             // stride1[47:16]

        v4i g2 = {0, 0, 0, 0};     // 2D tensor: groups 2/3 unused
        v4i g3 = {0, 0, 0, 0};
        v8i g4 = {0, 0, 0, 0, 0, 0, 0, 0};   // extra operand on clang-23 6-arg form

        __builtin_amdgcn_tensor_load_to_lds(g0, g1, g2, g3, g4, 0);
        __builtin_amdgcn_s_wait_tensorcnt(0);
    }
    asm volatile("" ::: "memory");
    __syncthreads();

    // ---------------- bin reduction out of LDS ----------------
    if (tid < CH * PH * PW) {
        int cl  = tid / (PH * PW);
        int bin = tid % (PH * PW);
        int i   = bin / PW;
        int j   = bin % PW;

        int hs = imin(imax((int)floorf((float)i * bsh) + oy1, 0), H_DIM);
        int he = imin(imax((int)ceilf(((float)i + 1.0f) * bsh) + oy1, 0), H_DIM);
        int ws = imin(imax((int)floorf((float)j * bsw) + ox1, 0), W_DIM);
        int we = imin(imax((int)ceilf(((float)j + 1.0f) * bsw) + ox1, 0), W_DIM);

        const float* tile = &s_tile[cl * nElems];
        float m = -__builtin_inff();
        for (int h = hs; h < he; ++h) {
            bool hin = (h > iy1) & (h < iy2);
            int rb = (h - h0) * W_DIM;
            for (int w = ws; w < we; ++w) {
                bool win = (w > ix1) & (w < ix2);
                if (!(hin & win)) m = fmaxf(m, tile[rb + w]);
            }
        }
        float res = __builtin_isfinite(m) ? m : 0.0f;   // empty ring -> 0
        out[((size_t)r * C_DIM + (size_t)(c0 + cl)) * (PH * PW) + bin] = res;
    }
}

extern "C" void kernel_launch(void* const* d_in, const int* in_sizes, int n_in,
                              void* d_out, int out_size, void* d_ws, size_t ws_size,
                              hipStream_t stream) {
    const float* feat = (const float*)d_in[0];   // (4, 256, 50, 50) f32
    const float* rois = (const float*)d_in[1];   // (128, 5) f32
    float* out = (float*)d_out;                  // (128, 256, 7, 7) f32
    (void)in_sizes; (void)n_in; (void)out_size; (void)d_ws; (void)ws_size;

    dim3 grid(R_DIM, C_DIM / CH, 1);   // (128, 64)
    dim3 block(256, 1, 1);             // 8 waves (wave32)
    roi_ring_pool_kernel<<<grid, block, 0, stream>>>(feat, rois, out);
}